// HANLayer_56573309223580
// MI455X (gfx1250) — compile-verified
//
#include <hip/hip_runtime.h>
#include <hip/hip_bf16.h>

typedef float v2f __attribute__((ext_vector_type(2)));
typedef float v8f __attribute__((ext_vector_type(8)));

#define CIN 128
#define COUT 128
#define HD 8     // heads
#define DD 16    // head dim

// ---------------------------------------------------------------------------
// init: zero the scatter accumulator + segment max/sum arrays
// ---------------------------------------------------------------------------
__global__ void han_init_ws(float* __restrict__ outacc, size_t nc,
                            unsigned int* __restrict__ mEnc, float* __restrict__ ssum) {
    size_t stride = (size_t)gridDim.x * blockDim.x;
    size_t i0 = (size_t)blockIdx.x * blockDim.x + threadIdx.x;
    for (size_t i = i0; i < nc; i += stride) outacc[i] = 0.0f;
    if (i0 < 128) { mEnc[i0] = 0u; ssum[i0] = 0.0f; }
}

// ---------------------------------------------------------------------------
// alpha = softmax(semantic_att, axis=0)  -> flattened [H*D] = [COUT]
// ---------------------------------------------------------------------------
__global__ void han_alpha(const float* __restrict__ sa, float* __restrict__ alpha) {
    int d = threadIdx.x;
    if (d >= DD) return;
    float m = -1e30f;
    for (int h = 0; h < HD; ++h) m = fmaxf(m, sa[h * DD + d]);
    float s = 0.0f;
    for (int h = 0; h < HD; ++h) s += __expf(sa[h * DD + d] - m);
    for (int h = 0; h < HD; ++h) alpha[h * DD + d] = __expf(sa[h * DD + d] - m) / s;
}

// ---------------------------------------------------------------------------
// QKV projection GEMM via V_WMMA_F32_16X16X4_F32.
// One wave computes one 16x16 tile of q, k, or v:  out = x @ W^T + b.
// A tile (16xK of x), B tile (Kx16 of W^T, i.e. W rows).
// ---------------------------------------------------------------------------
__global__ void han_qkv_gemm(const float* __restrict__ x,
                             const float* __restrict__ Wq, const float* __restrict__ bq,
                             const float* __restrict__ Wk, const float* __restrict__ bk,
                             const float* __restrict__ Wv, const float* __restrict__ bv,
                             float* __restrict__ qb, float* __restrict__ kb, float* __restrict__ vb,
                             int N) {
    int wave = (int)((blockIdx.x * blockDim.x + threadIdx.x) >> 5);
    int lane = threadIdx.x & 31;
    int mtiles = (N + 15) >> 4;
    int mt = wave % mtiles;
    int rest = wave / mtiles;
    int ct = rest & 7;          // 8 column tiles of 16 -> COUT=128
    int which = rest >> 3;      // 0=q, 1=k, 2=v
    if (which >= 3) return;     // wave-uniform: EXEC stays full for WMMA

    const float* W; const float* b; float* out;
    if (which == 0)      { W = Wq; b = bq; out = qb; }
    else if (which == 1) { W = Wk; b = bk; out = kb; }
    else                 { W = Wv; b = bv; out = vb; }

    int hl  = lane >> 4;        // half: 0 -> K=k0,k0+1 ; 1 -> K=k0+2,k0+3
    int l16 = lane & 15;
    int row  = mt * 16 + l16;
    int srow = row < N ? row : (N - 1);           // clamp (keep EXEC all-1s)
    const float* xrow = x + (size_t)srow * CIN;
    const float* wcol = W + (size_t)(ct * 16 + l16) * CIN;   // W row = output column
    float bval = b[ct * 16 + l16];

    v8f acc;
#pragma unroll
    for (int i = 0; i < 8; ++i) acc[i] = bval;

#pragma unroll
    for (int k0 = 0; k0 < CIN; k0 += 4) {
        v2f a  = *(const v2f*)(xrow + k0 + 2 * hl);
        v2f bb = *(const v2f*)(wcol + k0 + 2 * hl);
        acc = __builtin_amdgcn_wmma_f32_16x16x4_f32(false, a, false, bb,
                                                    (short)0, acc, false, false);
    }

    int baserow = mt * 16 + hl * 8;
    int colc = ct * 16 + l16;
#pragma unroll
    for (int i = 0; i < 8; ++i) {
        int r = baserow + i;
        if (r < N) out[(size_t)r * COUT + colc] = acc[i];
    }
}

// ---------------------------------------------------------------------------
// Edge scores: one wave per edge. lane l covers channels [4l, 4l+4) -> head l/4.
// score[h] = dot(q[row],k[col]) over head h / sqrt(D); segment max via
// ordered-int atomicMax encoding.
// ---------------------------------------------------------------------------
__global__ void han_edge_scores(const float* __restrict__ qb, const float* __restrict__ kb,
                                const int* __restrict__ rowp, const int* __restrict__ colp,
                                const int* __restrict__ etype,
                                float* __restrict__ scores, unsigned int* __restrict__ mEnc,
                                int E) {
    int e = (int)((blockIdx.x * blockDim.x + threadIdx.x) >> 5);
    int lane = threadIdx.x & 31;
    if (e >= E) return;
    int r = rowp[e], c = colp[e];
    float4 qv = *(const float4*)(qb + (size_t)r * COUT + lane * 4);
    float4 kv = *(const float4*)(kb + (size_t)c * COUT + lane * 4);
    float p = qv.x * kv.x + qv.y * kv.y + qv.z * kv.z + qv.w * kv.w;
    p += __shfl_xor(p, 1, 32);
    p += __shfl_xor(p, 2, 32);
    if ((lane & 3) == 0) {
        float s = p * 0.25f;                      // 1/sqrt(16)
        int h = lane >> 2;
        scores[(size_t)e * HD + h] = s;
        unsigned int u = __float_as_uint(s);
        u = (u & 0x80000000u) ? ~u : (u | 0x80000000u);
        atomicMax(&mEnc[etype[e] * HD + h], u);
    }
}

// ---------------------------------------------------------------------------
// e = exp(score - m[rel]); segment sum; store e in place of score.
// ---------------------------------------------------------------------------
__global__ void han_edge_exp(float* __restrict__ scores, const int* __restrict__ etype,
                             const unsigned int* __restrict__ mEnc, float* __restrict__ ssum,
                             int E) {
    long long i = (long long)blockIdx.x * blockDim.x + threadIdx.x;
    if (i >= (long long)E * HD) return;
    int e = (int)(i >> 3), h = (int)(i & 7);
    unsigned int enc = mEnc[etype[e] * HD + h];
    unsigned int bits = (enc & 0x80000000u) ? (enc & 0x7FFFFFFFu) : ~enc;
    float m = __uint_as_float(bits);
    float ev = __expf(scores[i] - m);
    scores[i] = ev;
    atomicAdd(&ssum[etype[e] * HD + h], ev);
}

// ---------------------------------------------------------------------------
// Scatter: one wave per edge; msg = v[col] * attn[h]; atomic add into out[row].
// ---------------------------------------------------------------------------
__global__ void han_edge_scatter(const float* __restrict__ vbuf, const float* __restrict__ scores,
                                 const float* __restrict__ ssum,
                                 const int* __restrict__ rowp, const int* __restrict__ colp,
                                 const int* __restrict__ etype,
                                 float* __restrict__ outacc, int E) {
    int e = (int)((blockIdx.x * blockDim.x + threadIdx.x) >> 5);
    int lane = threadIdx.x & 31;
    if (e >= E) return;
    int r = rowp[e], c = colp[e], t = etype[e];
    int h = lane >> 2;
    float attn = scores[(size_t)e * HD + h] / ssum[t * HD + h];
    float4 vv = *(const float4*)(vbuf + (size_t)c * COUT + lane * 4);
    float* op = outacc + (size_t)r * COUT + lane * 4;
    atomicAdd(op + 0, vv.x * attn);
    atomicAdd(op + 1, vv.y * attn);
    atomicAdd(op + 2, vv.z * attn);
    atomicAdd(op + 3, vv.w * attn);
}

// ---------------------------------------------------------------------------
// Final GEMM: out = (outacc * alpha[col]) @ Wo^T + bo, via WMMA f32 16x16x4.
// ---------------------------------------------------------------------------
__global__ void han_out_gemm(const float* __restrict__ g, const float* __restrict__ alpha,
                             const float* __restrict__ Wo, const float* __restrict__ bo,
                             float* __restrict__ out, int N) {
    int wave = (int)((blockIdx.x * blockDim.x + threadIdx.x) >> 5);
    int lane = threadIdx.x & 31;
    int mtiles = (N + 15) >> 4;
    int mt = wave % mtiles;
    int ct = wave / mtiles;
    if (ct >= 8) return;        // wave-uniform

    int hl  = lane >> 4;
    int l16 = lane & 15;
    int row  = mt * 16 + l16;
    int srow = row < N ? row : (N - 1);
    const float* grow = g  + (size_t)srow * COUT;
    const float* wcol = Wo + (size_t)(ct * 16 + l16) * COUT;
    float bval = bo[ct * 16 + l16];

    v8f acc;
#pragma unroll
    for (int i = 0; i < 8; ++i) acc[i] = bval;

#pragma unroll
    for (int k0 = 0; k0 < COUT; k0 += 4) {
        v2f a, bb;
        a.x = grow[k0 + 2 * hl]     * alpha[k0 + 2 * hl];
        a.y = grow[k0 + 2 * hl + 1] * alpha[k0 + 2 * hl + 1];
        bb = *(const v2f*)(wcol + k0 + 2 * hl);
        acc = __builtin_amdgcn_wmma_f32_16x16x4_f32(false, a, false, bb,
                                                    (short)0, acc, false, false);
    }

    int baserow = mt * 16 + hl * 8;
    int colc = ct * 16 + l16;
#pragma unroll
    for (int i = 0; i < 8; ++i) {
        int r = baserow + i;
        if (r < N) out[(size_t)r * COUT + colc] = acc[i];
    }
}

// ---------------------------------------------------------------------------
extern "C" void kernel_launch(void* const* d_in, const int* in_sizes, int n_in,
                              void* d_out, int out_size, void* d_ws, size_t ws_size,
                              hipStream_t stream) {
    const float* x        = (const float*)d_in[0];
    const int*   eidx     = (const int*)d_in[1];
    const int*   etype    = (const int*)d_in[2];
    // d_in[3] = num_relations (R=5; segment arrays padded to 16*8, not needed here)
    const float* Wq = (const float*)d_in[4];  const float* bq = (const float*)d_in[5];
    const float* Wk = (const float*)d_in[6];  const float* bk = (const float*)d_in[7];
    const float* Wv = (const float*)d_in[8];  const float* bv = (const float*)d_in[9];
    const float* sa = (const float*)d_in[10];
    const float* Wo = (const float*)d_in[11]; const float* bo = (const float*)d_in[12];
    float* out = (float*)d_out;

    int N = in_sizes[0] / CIN;
    int E = in_sizes[2];
    const int* rowp = eidx;
    const int* colp = eidx + E;

    // workspace layout (floats)
    size_t nc = (size_t)N * COUT;
    float* ws      = (float*)d_ws;
    float* qb      = ws;
    float* kb      = qb + nc;
    float* vb      = kb + nc;
    float* scores  = vb + nc;                    // E * 8
    float* outacc  = scores + (size_t)E * HD;    // N * 128
    unsigned int* mEnc = (unsigned int*)(outacc + nc);   // 128
    float* ssum    = (float*)(mEnc + 128);               // 128
    float* alpha   = ssum + 128;                         // 128

    // 1) zero accumulator + segment stats
    han_init_ws<<<1024, 256, 0, stream>>>(outacc, nc, mEnc, ssum);

    // 2) alpha = softmax over heads of semantic_att
    han_alpha<<<1, 32, 0, stream>>>(sa, alpha);

    // 3) q,k,v projections (WMMA)
    int mtiles = (N + 15) >> 4;
    {
        long long waves = 3LL * 8 * mtiles;
        int blocks = (int)((waves + 7) / 8);
        han_qkv_gemm<<<blocks, 256, 0, stream>>>(x, Wq, bq, Wk, bk, Wv, bv,
                                                 qb, kb, vb, N);
    }

    // 4) per-edge scores + per-relation max
    {
        int blocks = (E + 7) / 8;   // 8 waves (edges) per 256-thread block
        han_edge_scores<<<blocks, 256, 0, stream>>>(qb, kb, rowp, colp, etype,
                                                    scores, mEnc, E);
    }

    // 5) exp + per-relation sum
    {
        long long total = (long long)E * HD;
        int blocks = (int)((total + 255) / 256);
        han_edge_exp<<<blocks, 256, 0, stream>>>(scores, etype, mEnc, ssum, E);
    }

    // 6) weighted scatter-add of messages
    {
        int blocks = (E + 7) / 8;
        han_edge_scatter<<<blocks, 256, 0, stream>>>(vb, scores, ssum,
                                                     rowp, colp, etype, outacc, E);
    }

    // 7) output projection with alpha scaling (WMMA) -> d_out
    {
        long long waves = 8LL * mtiles;
        int blocks = (int)((waves + 7) / 8);
        han_out_gemm<<<blocks, 256, 0, stream>>>(outacc, alpha, Wo, bo, out, N);
    }
}